// DrainageGNN_59665685676525
// MI455X (gfx1250) — compile-verified
//
#include <hip/hip_runtime.h>

typedef __attribute__((ext_vector_type(2))) float v2f;
typedef __attribute__((ext_vector_type(8))) float v8f;

#define HIDN 64

// ---------------------------------------------------------------- utilities
__global__ __launch_bounds__(256) void zero_kernel(float* __restrict__ p, long n) {
    long i = (long)blockIdx.x * blockDim.x + threadIdx.x;
    if (i < n) p[i] = 0.0f;
}

__global__ __launch_bounds__(256) void deg_kernel(const int* __restrict__ dst,
                                                  float* __restrict__ cnt, int E) {
    int e = blockIdx.x * blockDim.x + threadIdx.x;
    if (e < E)
        __hip_atomic_fetch_add(&cnt[dst[e]], 1.0f, __ATOMIC_RELAXED, __HIP_MEMORY_SCOPE_AGENT);
}

__global__ __launch_bounds__(256) void inv_kernel(float* __restrict__ cnt, int n) {
    int i = blockIdx.x * blockDim.x + threadIdx.x;
    if (i < n) {
        float c = cnt[i];
        cnt[i] = (c > 0.0f) ? (1.0f / c) : 0.0f;
    }
}

// ------------------------------------------------- edge gather + atomic scatter
// one thread = one edge x 4 contiguous channels (float4 gather, 4x atomic_add_f32)
__global__ __launch_bounds__(256) void scatter_kernel(const float* __restrict__ h,
                                                      const int* __restrict__ src,
                                                      const int* __restrict__ dst,
                                                      float* __restrict__ agg,
                                                      int E, int d) {
    const int dvec = d >> 2;
    long tid = (long)blockIdx.x * blockDim.x + threadIdx.x;
    long total = (long)E * dvec;
    if (tid >= total) return;
    int e  = (int)(tid / dvec);
    int c4 = (int)(tid % dvec) << 2;
    int s = src[e], t = dst[e];
    const float4 v = *(const float4*)(h + (size_t)s * d + c4);
    float* p = agg + (size_t)t * d + c4;
    __hip_atomic_fetch_add(p + 0, v.x, __ATOMIC_RELAXED, __HIP_MEMORY_SCOPE_AGENT);
    __hip_atomic_fetch_add(p + 1, v.y, __ATOMIC_RELAXED, __HIP_MEMORY_SCOPE_AGENT);
    __hip_atomic_fetch_add(p + 2, v.z, __ATOMIC_RELAXED, __HIP_MEMORY_SCOPE_AGENT);
    __hip_atomic_fetch_add(p + 3, v.w, __ATOMIC_RELAXED, __HIP_MEMORY_SCOPE_AGENT);
}

// ------------------------------------------------------------- WMMA SAGE GEMM
// out[n,64] = (A .* inv) @ Wn[K,64] + bn + H @ Wr[K,64]   (relu optional)
// One wave computes one 16x16 output tile via V_WMMA_F32_16X16X4_F32 (exact fp32).
// A-fragment (16x4 f32): lanes 0-15 -> K+{0,1}, lanes 16-31 -> K+{2,3}.
// C/D: VGPR j -> row m0 + j + 8*(lane>=16), col n0 + (lane&15).
__global__ __launch_bounds__(256) void sage_gemm_kernel(
    const float* __restrict__ A,  const float* __restrict__ inv,
    const float* __restrict__ Wn, const float* __restrict__ bn,
    const float* __restrict__ H,  const float* __restrict__ Wr,
    float* __restrict__ out, int n, int K, int relu)
{
    int wave = threadIdx.x >> 5;
    int lane = threadIdx.x & 31;
    int tile = blockIdx.x * (blockDim.x >> 5) + wave;
    int mtiles = (n + 15) >> 4;
    int ntiles = mtiles * (HIDN >> 4);
    if (tile >= ntiles) return;           // wave-uniform: EXEC stays all-ones for WMMA

    int mt = tile >> 2;                   // HIDN/16 == 4 column tiles
    int nt = tile & 3;
    int m0 = mt << 4, n0 = nt << 4;
    int rc = lane & 15;                   // row within A-tile == col within B-tile
    int kk = (lane >> 4) << 1;            // 0 or 2
    int gm = m0 + rc; if (gm >= n) gm = n - 1;
    int gn = n0 + rc;

    v8f c = {0.f, 0.f, 0.f, 0.f, 0.f, 0.f, 0.f, 0.f};

    if (A) {
        float iv = inv ? inv[gm] : 1.0f;
        const float* arow = A + (size_t)gm * K;
        for (int k0 = 0; k0 < K; k0 += 4) {
            v2f a, b;
            a.x = arow[k0 + kk]     * iv;
            a.y = arow[k0 + kk + 1] * iv;
            b.x = Wn[(size_t)(k0 + kk)     * HIDN + gn];
            b.y = Wn[(size_t)(k0 + kk + 1) * HIDN + gn];
            c = __builtin_amdgcn_wmma_f32_16x16x4_f32(false, a, false, b,
                                                      (short)0, c, false, false);
        }
    }
    if (H) {
        const float* hrow = H + (size_t)gm * K;
        for (int k0 = 0; k0 < K; k0 += 4) {
            v2f a, b;
            a.x = hrow[k0 + kk];
            a.y = hrow[k0 + kk + 1];
            b.x = Wr[(size_t)(k0 + kk)     * HIDN + gn];
            b.y = Wr[(size_t)(k0 + kk + 1) * HIDN + gn];
            c = __builtin_amdgcn_wmma_f32_16x16x4_f32(false, a, false, b,
                                                      (short)0, c, false, false);
        }
    }

    float bias = bn[gn];
    int mbase = m0 + ((lane >> 4) << 3);
#pragma unroll
    for (int j = 0; j < 8; ++j) {
        int m = mbase + j;
        float v = c[j] + bias;
        if (relu) v = fmaxf(v, 0.0f);
        if (m < n) out[(size_t)m * HIDN + gn] = v;
    }
}

// ------------------------------------------------- final 64->1 dot + sigmoid
__global__ __launch_bounds__(256) void final_kernel(const float* __restrict__ h,
                                                    const float* __restrict__ Wp2,
                                                    const float* __restrict__ bp2,
                                                    float* __restrict__ out, int n) {
    int i = blockIdx.x * blockDim.x + threadIdx.x;
    if (i >= n) return;
    const float4* hp = (const float4*)(h + (size_t)i * HIDN);
    const float4* wp = (const float4*)Wp2;
    float s = bp2[0];
#pragma unroll
    for (int j = 0; j < HIDN / 4; ++j) {
        float4 a = hp[j], b = wp[j];
        s += a.x * b.x + a.y * b.y + a.z * b.z + a.w * b.w;
    }
    out[i] = 1.0f / (1.0f + expf(-s));
}

// ---------------------------------------------------------------------------
extern "C" void kernel_launch(void* const* d_in, const int* in_sizes, int n_in,
                              void* d_out, int out_size, void* d_ws, size_t ws_size,
                              hipStream_t stream) {
    const float* x   = (const float*)d_in[0];
    const int*   ei  = (const int*)  d_in[1];
    const float* Wn0 = (const float*)d_in[2];
    const float* bn0 = (const float*)d_in[3];
    const float* Wr0 = (const float*)d_in[4];
    const float* Wn1 = (const float*)d_in[5];
    const float* bn1 = (const float*)d_in[6];
    const float* Wr1 = (const float*)d_in[7];
    const float* Wn2 = (const float*)d_in[8];
    const float* bn2 = (const float*)d_in[9];
    const float* Wr2 = (const float*)d_in[10];
    const float* Wp1 = (const float*)d_in[11];
    const float* bp1 = (const float*)d_in[12];
    const float* Wp2 = (const float*)d_in[13];
    const float* bp2 = (const float*)d_in[14];

    const int IN = 32;
    const int n  = in_sizes[0] / IN;
    const int E  = in_sizes[1] / 2;
    const int* src = ei;
    const int* dst = ei + E;

    float* inv  = (float*)d_ws;                 // [n]
    float* agg  = inv  + n;                     // [n*64]
    float* bufA = agg  + (size_t)n * HIDN;      // [n*64]
    float* bufB = bufA + (size_t)n * HIDN;      // [n*64]

    auto cdiv = [](long a, long b) { return (int)((a + b - 1) / b); };

    // ---- in-degree normalization
    zero_kernel<<<cdiv(n, 256), 256, 0, stream>>>(inv, (long)n);
    deg_kernel <<<cdiv(E, 256), 256, 0, stream>>>(dst, inv, E);
    inv_kernel <<<cdiv(n, 256), 256, 0, stream>>>(inv, n);

    int mtiles  = cdiv(n, 16);
    int gtiles  = mtiles * (HIDN / 16);
    int gblocks = cdiv(gtiles, 8);              // 8 waves (wave32) per 256-thread block

    // ---- SAGE layer 0 (K=32)
    zero_kernel   <<<cdiv((long)n * IN, 256), 256, 0, stream>>>(agg, (long)n * IN);
    scatter_kernel<<<cdiv((long)E * (IN / 4), 256), 256, 0, stream>>>(x, src, dst, agg, E, IN);
    sage_gemm_kernel<<<gblocks, 256, 0, stream>>>(agg, inv, Wn0, bn0, x, Wr0, bufA, n, IN, 1);

    // ---- SAGE layer 1 (K=64)
    zero_kernel   <<<cdiv((long)n * HIDN, 256), 256, 0, stream>>>(agg, (long)n * HIDN);
    scatter_kernel<<<cdiv((long)E * (HIDN / 4), 256), 256, 0, stream>>>(bufA, src, dst, agg, E, HIDN);
    sage_gemm_kernel<<<gblocks, 256, 0, stream>>>(agg, inv, Wn1, bn1, bufA, Wr1, bufB, n, HIDN, 1);

    // ---- SAGE layer 2 (K=64, no activation)
    zero_kernel   <<<cdiv((long)n * HIDN, 256), 256, 0, stream>>>(agg, (long)n * HIDN);
    scatter_kernel<<<cdiv((long)E * (HIDN / 4), 256), 256, 0, stream>>>(bufB, src, dst, agg, E, HIDN);
    sage_gemm_kernel<<<gblocks, 256, 0, stream>>>(agg, inv, Wn2, bn2, bufB, Wr2, bufA, n, HIDN, 0);

    // ---- predictor hidden layer: relu(bufA @ Wp1 + bp1)
    sage_gemm_kernel<<<gblocks, 256, 0, stream>>>(bufA, nullptr, Wp1, bp1,
                                                  nullptr, nullptr, bufB, n, HIDN, 1);

    // ---- final 64->1 + sigmoid
    final_kernel<<<cdiv(n, 256), 256, 0, stream>>>(bufB, Wp2, bp2, (float*)d_out, n);

    (void)n_in; (void)out_size; (void)ws_size;
}